// KnnEstimator_17154099381057
// MI455X (gfx1250) — compile-verified
//
#include <hip/hip_runtime.h>

#define K_OUT   50
#define NQ      4096
#define NOBS    65536
#define QTILE   16          // queries per workgroup (one WMMA tile of rows)
#define THREADS 256         // 8 waves (wave32)
#define WCOLS   64          // obs columns per wave per iteration (4 WMMA tiles)
#define COLS    (8 * WCOLS) // obs columns per workgroup per iteration = 512
#define ITERS   (NOBS / COLS)
#define EARTH_R 6371000.0f

typedef float v2f __attribute__((ext_vector_type(2)));
typedef float v8f __attribute__((ext_vector_type(8)));

// ---------------------------------------------------------------------------
// Phase 1: (lat,lng) -> unit vector u = (cos(lat)cos(lng), cos(lat)sin(lng),
// sin(lat)).  cos(theta) = dot(uq,uo), haversine a = (1-dot)/2, so distance
// ranking == dot ranking (reversed) and the matrix is a K=3 (pad 4) GEMM.
// Vectors are stored pre-swizzled into WMMA fragment halves:
//   frag[2*i+0] = (x, y)   <- K={0,1} half (lanes 0-15)
//   frag[2*i+1] = (z, 0)   <- K={2,3} half (lanes 16-31)
// so the GEMM loop needs zero per-load lane selects.
// ---------------------------------------------------------------------------
__global__ void prep_vectors(const float* __restrict__ qcoord,
                             const float* __restrict__ ocoord,
                             float2* __restrict__ qfrag,
                             float2* __restrict__ ofrag)
{
    int i = blockIdx.x * blockDim.x + threadIdx.x;
    if (i < NOBS) {
        float lat = ocoord[2 * i + 0];
        float lng = ocoord[2 * i + 1];
        float sl, cl, sg, cg;
        sincosf(lat, &sl, &cl);
        sincosf(lng, &sg, &cg);
        ofrag[2 * i + 0] = make_float2(cl * cg, cl * sg);
        ofrag[2 * i + 1] = make_float2(sl, 0.0f);
    } else if (i < NOBS + NQ) {
        int j = i - NOBS;
        float lat = qcoord[2 * j + 0];
        float lng = qcoord[2 * j + 1];
        float sl, cl, sg, cg;
        sincosf(lat, &sl, &cl);
        sincosf(lng, &sg, &cg);
        qfrag[2 * j + 0] = make_float2(cl * cg, cl * sg);
        qfrag[2 * j + 1] = make_float2(sl, 0.0f);
    }
}

// ---------------------------------------------------------------------------
// Phase 2: one workgroup = 16 queries, 8 waves, no barriers in the main loop.
// Each wave: 4x V_WMMA_F32_16X16X4_F32 per iteration into a wave-private LDS
// tile, then its 32 lanes (2 per query, each owning a contiguous 32-column
// half-row -> ds_load_b128 scans) maintain per-lane top-50-by-dot lists.
// Same-wave LDS ops are in-order, so produce->consume needs no barrier.
// ---------------------------------------------------------------------------
__global__ __launch_bounds__(THREADS)
void knn_wmma(const float2* __restrict__ qfrag,
              const float2* __restrict__ ofrag,
              float* __restrict__ out_dist,
              int*   __restrict__ out_idx)
{
    extern __shared__ char smem_raw[];
    float* tiles   = (float*)smem_raw;                      // 8 * QTILE * WCOLS
    float* lists_v = tiles + 8 * QTILE * WCOLS;             // THREADS * K_OUT
    int*   lists_i = (int*)(lists_v + THREADS * K_OUT);     // THREADS * K_OUT
    float* red_v   = (float*)(lists_i + THREADS * K_OUT);   // THREADS
    int*   red_i   = (int*)(red_v + THREADS);               // THREADS
    int*   win     = (int*)(red_i + THREADS);               // QTILE

    const int tid    = threadIdx.x;
    const int wave   = tid >> 5;
    const int lane   = tid & 31;
    const int laneN  = lane & 15;
    const int laneHi = lane >> 4;          // 0: K={0,1} half, 1: K={2,3} half

    const int qBase = blockIdx.x * QTILE;

    // A fragment: pre-swizzled load, no cndmask needed.
    float2 qf = qfrag[2 * (qBase + laneN) + laneHi];
    v2f afrag; afrag.x = qf.x; afrag.y = qf.y;

    // Per-lane base for B-fragment loads (folds laneHi into the pointer).
    const float2* obase = ofrag + laneHi;

    // Per-lane top-K list (LDS), keyed on dot (keep the K largest).
    float* lv = lists_v + tid * K_OUT;
    int*   li = lists_i + tid * K_OUT;
    for (int i = 0; i < K_OUT; ++i) { lv[i] = -__builtin_inff(); li[i] = -1; }
    float curMin = -__builtin_inff();
    int   curPos = 0;

    float* wtile = tiles + wave * (QTILE * WCOLS);  // wave-private tile

    // Selection roles: 2 lanes per query, each owns a contiguous half-row.
    const int mq = lane >> 1;   // query row 0..15 owned by this lane
    const int mh = lane & 1;    // half-row: columns [mh*32, mh*32+32)

    for (int it = 0; it < ITERS; ++it) {
        const int stripe = it * COLS + wave * WCOLS;   // this wave's 64 obs

        // Prefetch next stripe (global_prefetch_b8); ofrag is 1 MB -> L2-hot.
        if (it + 1 < ITERS) {
            __builtin_prefetch(obase + 2 * (stripe + COLS) + lane * 4, 0, 1);
        }

        // ----- compute: four 16x16 WMMA dot tiles per wave -----
        #pragma unroll
        for (int g = 0; g < WCOLS / 16; ++g) {
            const int col = g * 16 + laneN;            // 0..63 within stripe
            float2 of = obase[2 * (stripe + col)];     // global_load_b64
            v2f bfrag; bfrag.x = of.x; bfrag.y = of.y;

            v8f acc = {};
            acc = __builtin_amdgcn_wmma_f32_16x16x4_f32(
                      false, afrag, false, bfrag, (short)0, acc, false, false);

            // C/D layout: VGPR r -> row M=r (lanes 0-15) / M=r+8 (lanes 16-31)
            #pragma unroll
            for (int r = 0; r < 8; ++r)
                wtile[(r + 8 * laneHi) * WCOLS + col] = acc[r];
        }
        // Same-wave LDS ordering: no barrier before consuming wtile.

        // ----- select: scan own half-row with b128 loads, keep max-dot-50 ---
        const float4* wrow = (const float4*)(wtile + mq * WCOLS + mh * 32);
        const int idxBase  = stripe + mh * 32;
        #pragma unroll
        for (int j = 0; j < 8; ++j) {
            const float4 d4 = wrow[j];                 // ds_load_b128
            const float d[4] = { d4.x, d4.y, d4.z, d4.w };
            #pragma unroll
            for (int e = 0; e < 4; ++e) {
                if (d[e] > curMin) {
                    lv[curPos] = d[e];
                    li[curPos] = idxBase + 4 * j + e;
                    curMin = __builtin_inff();
                    for (int i = 0; i < K_OUT; ++i)
                        if (lv[i] < curMin) { curMin = lv[i]; curPos = i; }
                }
            }
        }
    }

    __syncthreads();   // only workgroup-wide sync before the merge

    // ----- merge: per query, 16 candidate lists (8 waves x 2 half-rows) -----
    const int sub = (wave << 1) | mh;      // merge sub-id 0..15 within query
    for (int k = 0; k < K_OUT; ++k) {
        float mv = -__builtin_inff();
        int   mi = -1, mp = 0;
        for (int i = 0; i < K_OUT; ++i)
            if (lv[i] > mv) { mv = lv[i]; mi = li[i]; mp = i; }
        red_v[tid] = mv;
        red_i[tid] = mi;
        __syncthreads();
        if (sub == 0) {                    // leader: wave 0, even lane
            float best = -__builtin_inff();
            int bs = 0, bi = -1;
            for (int s = 0; s < 16; ++s) {
                const int ts = (s >> 1) * 32 + mq * 2 + (s & 1);
                const float v = red_v[ts];
                if (v > best) { best = v; bs = s; bi = red_i[ts]; }
            }
            // Largest dot == smallest distance; convert only here.
            const float aa   = fminf(fmaxf(0.5f - 0.5f * best, 0.0f), 1.0f);
            const float dist = 2.0f * asinf(sqrtf(aa)) * EARTH_R;
            const int   qg   = qBase + mq;
            out_dist[qg * K_OUT + k] = dist;
            out_idx [qg * K_OUT + k] = bi;
            win[mq] = bs;
        }
        __syncthreads();
        if (sub == win[mq]) lv[mp] = -__builtin_inff();   // consume winner
        __syncthreads();
    }
}

// ---------------------------------------------------------------------------
extern "C" void kernel_launch(void* const* d_in, const int* in_sizes, int n_in,
                              void* d_out, int out_size, void* d_ws, size_t ws_size,
                              hipStream_t stream)
{
    const float* qcoord = (const float*)d_in[0];   // (4096, 2) f32
    const float* ocoord = (const float*)d_in[1];   // (65536, 2) f32

    float2* ofrag = (float2*)d_ws;                                         // 1 MB
    float2* qfrag = (float2*)((char*)d_ws + (size_t)2 * NOBS * sizeof(float2)); // 64 KB

    const int total = NOBS + NQ;
    prep_vectors<<<(total + 255) / 256, 256, 0, stream>>>(qcoord, ocoord, qfrag, ofrag);

    float* out_dist = (float*)d_out;               // first output: dists
    int*   out_idx  = (int*)d_out + NQ * K_OUT;    // second output: indices

    const size_t smem =
        (size_t)8 * QTILE * WCOLS * sizeof(float) + // wave-private dot tiles
        (size_t)THREADS * K_OUT * sizeof(float) +   // list values
        (size_t)THREADS * K_OUT * sizeof(int) +     // list indices
        (size_t)THREADS * sizeof(float) +           // reduction values
        (size_t)THREADS * sizeof(int) +             // reduction indices
        (size_t)QTILE * sizeof(int);                // winner slots

    knn_wmma<<<NQ / QTILE, THREADS, smem, stream>>>(qfrag, ofrag, out_dist, out_idx);
}